// MatrixAttention_44135083933705
// MI455X (gfx1250) — compile-verified
//
#include <hip/hip_runtime.h>

typedef __attribute__((ext_vector_type(2))) float v2f;
typedef __attribute__((ext_vector_type(4))) float v4f;
typedef __attribute__((ext_vector_type(8))) float v8f;

#define BATCH  8
#define N_ROWS 2048
#define D_DIM  512

// ---------------------------------------------------------------------------
// Phase 1: batched GEMV via V_WMMA_F32_16X16X4_F32 (full fp32 precision).
//   s[mat, b, n] = sum_d M[b,n,d] * w[mat*D + d]
// One wave32 handles one 16-row group. A-matrix = 16 rows x 4 d-values,
// B-matrix = w chunk broadcast across all 16 columns, so every column of the
// 16x16 C accumulator holds the same dot-product partial.
//
// ISA VGPR layouts (CDNA5 ISA):
//   A 16x4 f32 : lanes 0-15 -> M=lane, VGPR0=K0, VGPR1=K1; lanes 16-31 -> K2,K3
//   B 4x16 f32 : lanes 0-15 -> K rows {0,1}, lanes 16-31 -> K rows {2,3}
//   C 16x16 f32: lane 0 VGPR r = C[M=r][N=0]; lane 16 VGPR r = C[M=8+r][N=0]
// ---------------------------------------------------------------------------
__global__ __launch_bounds__(256) void gemv_wmma_f32(
    const float* __restrict__ m1, const float* __restrict__ m2,
    const float* __restrict__ w, float* __restrict__ s) {
  const int lane = threadIdx.x & 31;
  const int wave = threadIdx.x >> 5;
  const int g    = blockIdx.x * 8 + wave;   // 0..2047 row-groups
  const int mat  = g >> 10;                 // 0: matrix_1, 1: matrix_2
  const int rem  = g & 1023;
  const int b    = rem >> 7;                // batch index
  const int row0 = (rem & 127) * 16;        // first of 16 rows

  const float* __restrict__ M  = (mat == 0 ? m1 : m2) + (size_t)b * N_ROWS * D_DIM;
  const float* __restrict__ wv = w + mat * D_DIM;

  const int mrow = lane & 15;               // M row handled by this lane
  const int koff = (lane >> 4) * 2;         // K sub-offset: 0 for lanes<16, 2 else
  const float* __restrict__ arow = M + (size_t)(row0 + mrow) * D_DIM + koff;

  v8f c = {0.f, 0.f, 0.f, 0.f, 0.f, 0.f, 0.f, 0.f};
#pragma unroll 4
  for (int d0 = 0; d0 < D_DIM; d0 += 4) {
    v2f a  = *(const v2f*)(arow + d0);            // A[M=mrow][K=koff..koff+1]
    v2f bb = *(const v2f*)(wv + d0 + koff);       // w chunk, broadcast over N
    // 8 args: (neg_a, A, neg_b, B, c_mod, C, reuse_a, reuse_b)
    c = __builtin_amdgcn_wmma_f32_16x16x4_f32(
        false, a, false, bb, (short)0, c, false, false);
  }

  float* __restrict__ sout = s + mat * (BATCH * N_ROWS) + b * N_ROWS + row0;
  if (lane == 0) {
#pragma unroll
    for (int r = 0; r < 8; ++r) sout[r] = c[r];        // rows 0..7  (N=0 column)
  } else if (lane == 16) {
#pragma unroll
    for (int r = 0; r < 8; ++r) sout[8 + r] = c[r];    // rows 8..15 (N=0 column)
  }
}

// ---------------------------------------------------------------------------
// Phase 2: out[b,i,j] = s1[b,i] + s2[b,j] + bias.
// Store-bandwidth bound (134 MB): one float4 per thread, nontemporal b128
// stores so the streamed output does not thrash L2. s-rows (8 KB each) stay
// cache-resident across the i dimension.
// ---------------------------------------------------------------------------
__global__ __launch_bounds__(256) void broadcast_add(
    const float* __restrict__ s, const float* __restrict__ bias,
    float* __restrict__ out) {
  const int b = blockIdx.z;
  const int i = blockIdx.y;
  const int j = (blockIdx.x * 256 + threadIdx.x) * 4;

  const float s1v = s[b * N_ROWS + i] + bias[0];
  const v4f   s2v = *(const v4f*)(s + BATCH * N_ROWS + b * N_ROWS + j);
  v4f v = s2v + s1v;

  float* o = out + ((size_t)b * N_ROWS + i) * N_ROWS + j;
  __builtin_nontemporal_store(v, (v4f*)o);
}

extern "C" void kernel_launch(void* const* d_in, const int* in_sizes, int n_in,
                              void* d_out, int out_size, void* d_ws, size_t ws_size,
                              hipStream_t stream) {
  const float* m1   = (const float*)d_in[0];   // (8, 2048, 512) f32
  const float* m2   = (const float*)d_in[1];   // (8, 2048, 512) f32
  const float* w    = (const float*)d_in[2];   // (1024,) f32
  const float* bias = (const float*)d_in[3];   // (1,)   f32
  float*       out  = (float*)d_out;           // (8, 2048, 2048) f32
  float*       s    = (float*)d_ws;            // 2 * 8 * 2048 f32 = 128 KB

  // Phase 1: 2048 row-groups, 8 waves (one group each) per 256-thread block.
  gemv_wmma_f32<<<dim3(2048 / 8), dim3(256), 0, stream>>>(m1, m2, w, s);

  // Phase 2: grid (N2/1024, N1, B); each thread stores one float4.
  broadcast_add<<<dim3(N_ROWS / 1024, N_ROWS, BATCH), dim3(256), 0, stream>>>(
      s, bias, out);
}